// EMDLoss_40226663694452
// MI455X (gfx1250) — compile-verified
//
#include <hip/hip_runtime.h>
#include <hip/hip_bf16.h>

// EMD greedy-matching loss for MI455X (gfx1250, wave32, WMMA).
//
// Design: one workgroup per batch (B=8). Targets' B-fragments live in
// registers for the whole kernel. Rows are processed 16 at a time: the
// 16x2048 slab of s[i][j] = tt_j - 2*p_i.t_j is produced by
// V_WMMA_F32_16X16X4_F32 (rank-3 GEMM padded to K=4, full f32) and parked
// in a 128KB dynamic-LDS tile (CDNA5: 320KB LDS per workgroup). The 2048
// sequential greedy argmin steps then run entirely out of LDS/registers:
// packed (ordered-float<<32 | col) u64 keys, wave32 shuffle-min, one
// cross-wave LDS reduce, 2 barriers per step. sqrt/pp_i applied only to
// the winner (monotonicity), so HBM sees ~400KB total traffic.

typedef __attribute__((ext_vector_type(2))) float v2f;
typedef __attribute__((ext_vector_type(8))) float v8f;

#define BATCHES 8
#define NPTS    2048
#define BLOCK   512
#define NWAVES  (BLOCK / 32)            // 16
#define CPT     (NPTS / BLOCK)          // 4 columns per thread
#define TPW     ((NPTS / 16) / NWAVES)  // 8 column-tiles per wave
#define CHUNKS  (NPTS / 16)             // 128 row chunks

__global__ __launch_bounds__(BLOCK) void emd_greedy_kernel(
    const float* __restrict__ pred,
    const float* __restrict__ target,
    float* __restrict__ out)
{
    extern __shared__ char smem[];
    float* tile = (float*)smem;                               // [16][NPTS] = 128KB
    float* pps  = (float*)(smem + 16 * NPTS * sizeof(float)); // [NPTS]     = 8KB

    __shared__ unsigned           usedMask[NPTS / 32];
    __shared__ unsigned long long waveMin[NWAVES];
    __shared__ float              bsum;

    const int tid  = threadIdx.x;
    const int lane = tid & 31;
    const int wave = tid >> 5;
    const int b    = blockIdx.x;

    const float* __restrict__ P = pred   + (size_t)b * NPTS * 3;
    const float* __restrict__ T = target + (size_t)b * NPTS * 3;

    // ---- one-time setup --------------------------------------------------
    for (int k = tid; k < NPTS / 32; k += BLOCK) usedMask[k] = 0u;
    for (int i = tid; i < NPTS; i += BLOCK) {
        float x = P[i * 3 + 0], y = P[i * 3 + 1], z = P[i * 3 + 2];
        pps[i] = x * x + y * y + z * z;
    }
    if (tid == 0) bsum = 0.0f;

    // B fragments (4x16, K-padded with 0) + per-column tt, resident in
    // registers for the entire kernel. ISA layout: lanes 0-15 hold K=0/K=1
    // in VGPR0/1 for N=lane; lanes 16-31 hold K=2/K=3 for N=lane-16.
    v2f   Bf[TPW];
    float ttf[TPW];
#pragma unroll
    for (int t = 0; t < TPW; ++t) {
        int   c  = (wave * TPW + t) * 16 + (lane & 15);
        float tx = T[c * 3 + 0], ty = T[c * 3 + 1], tz = T[c * 3 + 2];
        ttf[t] = tx * tx + ty * ty + tz * tz;
        if (lane < 16) { Bf[t].x = tx; Bf[t].y = ty;   }
        else           { Bf[t].x = tz; Bf[t].y = 0.0f; }
    }
    __syncthreads();

    // ---- main loop: 128 chunks of 16 rows --------------------------------
    for (int chunk = 0; chunk < CHUNKS; ++chunk) {
        const int i0 = chunk * 16;

        // A fragment: rows i0..i0+15 of pred, pre-scaled by -2, K-padded.
        int r = i0 + (lane & 15);
        v2f Af;
        if (lane < 16) { Af.x = -2.0f * P[r * 3 + 0]; Af.y = -2.0f * P[r * 3 + 1]; }
        else           { Af.x = -2.0f * P[r * 3 + 2]; Af.y = 0.0f;                 }

        // 8 WMMAs per wave -> 16x2048 slab of (tt_j - 2 p.t) into LDS.
#pragma unroll
        for (int t = 0; t < TPW; ++t) {
            v8f C = {};
            C = __builtin_amdgcn_wmma_f32_16x16x4_f32(
                    false, Af, false, Bf[t], (short)0, C, false, false);
            int c     = (wave * TPW + t) * 16 + (lane & 15);
            int rbase = (lane < 16) ? 0 : 8;   // C VGPR r <-> row r (+8 for hi lanes)
#pragma unroll
            for (int rr = 0; rr < 8; ++rr)
                tile[(rbase + rr) * NPTS + c] = C[rr] + ttf[t];
        }
        __syncthreads();

        // 16 sequential greedy argmin steps over the LDS-resident slab.
        for (int step = 0; step < 16; ++step) {
            const float* __restrict__ rowp = tile + step * NPTS;
            const int j0 = tid * CPT;
            unsigned um  = usedMask[j0 >> 5] >> (j0 & 31); // our 4 used-bits

            unsigned long long best = ~0ull;
#pragma unroll
            for (int q = 0; q < CPT; ++q) {
                int      j = j0 + q;
                unsigned u = __float_as_uint(rowp[j]);
                u = (u & 0x80000000u) ? ~u : (u | 0x80000000u); // order-preserving
                unsigned long long key =
                    ((unsigned long long)u << 32) | (unsigned)j;
                if (!((um >> q) & 1u) && key < best) best = key;
            }
            // wave32 min-reduce (lowers to ds_swizzle/permute path)
#pragma unroll
            for (int off = 16; off > 0; off >>= 1) {
                unsigned long long o = __shfl_xor(best, off, 32);
                if (o < best) best = o;
            }
            if (lane == 0) waveMin[wave] = best;
            __syncthreads();

            if (wave == 0) {
                unsigned long long k = (lane < NWAVES) ? waveMin[lane] : ~0ull;
#pragma unroll
                for (int off = 16; off > 0; off >>= 1) {
                    unsigned long long o = __shfl_xor(k, off, 32);
                    if (o < k) k = o;
                }
                if (lane == 0) {
                    int jw = (int)(unsigned)k;
                    usedMask[jw >> 5] |= 1u << (jw & 31);   // single writer
                    unsigned u = (unsigned)(k >> 32);
                    float s = (u & 0x80000000u) ? __uint_as_float(u ^ 0x80000000u)
                                                : __uint_as_float(~u);
                    float d2 = s + pps[i0 + step];          // add row-constant |p|^2
                    bsum += sqrtf(fmaxf(d2, 1e-12f));
                }
            }
            __syncthreads();  // publish usedMask for next step / free tile reuse
        }
    }

    if (tid == 0)
        atomicAdd(out, bsum * (1.0f / (float)(BATCHES * NPTS)));
}

extern "C" void kernel_launch(void* const* d_in, const int* in_sizes, int n_in,
                              void* d_out, int out_size, void* d_ws, size_t ws_size,
                              hipStream_t stream)
{
    (void)in_sizes; (void)n_in; (void)out_size; (void)d_ws; (void)ws_size;
    const float* pred   = (const float*)d_in[0];
    const float* target = (const float*)d_in[1];
    float*       out    = (float*)d_out;

    hipMemsetAsync(out, 0, sizeof(float), stream);  // graph-capture safe

    const size_t shmem = (size_t)16 * NPTS * sizeof(float)  // cost slab (128KB)
                       + (size_t)NPTS * sizeof(float);      // |p_i|^2   (8KB)
    emd_greedy_kernel<<<dim3(BATCHES), dim3(BLOCK), shmem, stream>>>(pred, target, out);
}